// HeteroscedasticSoftmax_28698971472080
// MI455X (gfx1250) — compile-verified
//
#include <hip/hip_runtime.h>

// HeteroscedasticSoftmax on MI455X (gfx1250).
// Roofline: 120 MB HBM traffic (~5us @ 23.3 TB/s) vs ~2e10 VALU + ~3e9 TRANS
// ops -> purely compute-bound. One pixel per lane; the 19-channel softmax and
// accumulator live entirely in VGPRs. Staging of the 38 channel values
// (256 KB apart in memory) uses the CDNA5 async LDS-DMA path
// (global_load_async_to_lds_b32 + s_wait_asynccnt, ASYNCcnt-tracked).
// Hot loop uses the TRANS units natively: v_sin/v_cos in revolutions,
// v_log/v_exp in base-2 with log2(e) pre-folded into mu/sd.

#define CH       19
#define HW       65536        // 256*256
#define NPIX     524288       // 8*65536
#define NSAMP    100
#define BLK      256

__device__ __forceinline__ unsigned pcg_hash(unsigned v) {
    unsigned s = v * 747796405u + 2891336453u;
    unsigned w = ((s >> ((s >> 28u) + 4u)) ^ s) * 277803737u;
    return (w >> 22u) ^ w;
}

__global__ __launch_bounds__(BLK)
void hetero_softmax_kernel(const float* __restrict__ in, float* __restrict__ out) {
    __shared__ float tile[2 * CH * BLK];   // 38 KB staging buffer

    const int      tid = threadIdx.x;
    const unsigned g   = blockIdx.x * BLK + tid;   // global pixel id
    const unsigned b   = g >> 16;                  // batch index (g / 65536)
    const unsigned p   = g & 65535u;               // pixel within image

    const float* src = in + (size_t)b * (2 * CH) * HW + p;

    // ---- CDNA5 async DMA: gather 38 strided channel values into LDS ------
    // Each lane owns column `tid` of the tile; ASYNCcnt tracks completion.
    const unsigned lds_base = (unsigned)(size_t)(&tile[0]);
#pragma unroll
    for (int c = 0; c < 2 * CH; ++c) {
        unsigned long long ga = (unsigned long long)(src + (size_t)c * HW);
        unsigned           la = lds_base + (unsigned)((c * BLK + tid) * 4);
        asm volatile("global_load_async_to_lds_b32 %0, %1, off"
                     :: "v"(la), "v"(ga) : "memory");
    }
    asm volatile("s_wait_asynccnt 0" ::: "memory");
    // Each lane reads back only what it wrote -> no workgroup barrier needed.

    const float LOG2E = 1.4426950408889634f;

    // mu, sd pre-scaled by log2(e) so the softmax exponential is a bare
    // v_exp_f32 (2^x) with no per-channel multiply in the hot loop.
    float mu[CH], sd[CH];
#pragma unroll
    for (int c = 0; c < CH; ++c) mu[c] = tile[c * BLK + tid] * LOG2E;
#pragma unroll
    for (int c = 0; c < CH; ++c)
        sd[c] = __builtin_amdgcn_exp2f(tile[(CH + c) * BLK + tid] * LOG2E) * LOG2E;

    float acc[CH];
#pragma unroll
    for (int c = 0; c < CH; ++c) acc[c] = 0.0f;

    for (int s = 0; s < NSAMP; ++s) {
        const unsigned base = ((unsigned)s * (unsigned)NPIX + g) * 32u;

        float x[CH];
        // Counter-based Box-Muller: 10 pairs -> 20 normals (19 used).
        // v_sin/v_cos take revolutions, so u2 in [0,1) feeds them directly.
        // r = sqrt(-2 ln u1) = sqrt(-2ln2 * log2(u1)) via raw v_log_f32.
#pragma unroll
        for (int j = 0; j < (CH + 1) / 2; ++j) {
            unsigned h1 = pcg_hash(base + 2u * (unsigned)j);
            unsigned h2 = pcg_hash(base + 2u * (unsigned)j + 1u);
            float u1 = (float)((h1 >> 8) + 1u) * 5.9604644775390625e-8f; // (0,1]
            float u2 = (float)(h2 >> 8)        * 5.9604644775390625e-8f; // [0,1) rev
            float r  = __builtin_amdgcn_sqrtf(-1.3862943611198906f *
                                              __builtin_amdgcn_logf(u1));
            float z0 = r * __builtin_amdgcn_cosf(u2);
            float z1 = r * __builtin_amdgcn_sinf(u2);
            const int c0 = 2 * j, c1 = 2 * j + 1;
            x[c0] = __fmaf_rn(z0, sd[c0], mu[c0]);
            if (c1 < CH) x[c1] = __fmaf_rn(z1, sd[c1], mu[c1]);
        }

        // Numerically stable softmax over 19 channels (per lane, in VGPRs),
        // entirely in the exp2 domain.
        float xmax = x[0];
#pragma unroll
        for (int c = 1; c < CH; ++c) xmax = fmaxf(xmax, x[c]);
        float sum = 0.0f;
#pragma unroll
        for (int c = 0; c < CH; ++c) {
            float e = __builtin_amdgcn_exp2f(x[c] - xmax);
            x[c] = e;
            sum += e;
        }
        float rs = __builtin_amdgcn_rcpf(sum);
#pragma unroll
        for (int c = 0; c < CH; ++c) acc[c] = __fmaf_rn(x[c], rs, acc[c]);
    }

    float* dst = out + (size_t)b * CH * HW + p;
#pragma unroll
    for (int c = 0; c < CH; ++c) dst[(size_t)c * HW] = acc[c] * (1.0f / NSAMP);
}

extern "C" void kernel_launch(void* const* d_in, const int* in_sizes, int n_in,
                              void* d_out, int out_size, void* d_ws, size_t ws_size,
                              hipStream_t stream) {
    (void)in_sizes; (void)n_in; (void)out_size; (void)d_ws; (void)ws_size;
    const float* in  = (const float*)d_in[0];
    float*       out = (float*)d_out;
    hipLaunchKernelGGL(hetero_softmax_kernel,
                       dim3(NPIX / BLK), dim3(BLK), 0, stream, in, out);
}